// GATConv_24232205484631
// MI455X (gfx1250) — compile-verified
//
#include <hip/hip_runtime.h>

#define N_NODES 50000
#define N_EDGES 800000
#define IN_CH   128
#define HEADS   8
#define OUT_CH  32
#define HC      256            // HEADS*OUT_CH
#define E_TOT   (N_EDGES + N_NODES)
#define NEG_SLOPE 0.2f

typedef float v2f __attribute__((ext_vector_type(2)));
typedef float v8f __attribute__((ext_vector_type(8)));

// ---------- order-preserving float <-> uint mapping for atomic max ----------
__device__ __forceinline__ unsigned f2ord(float f) {
    unsigned u = __float_as_uint(f);
    return (u & 0x80000000u) ? ~u : (u | 0x80000000u);
}
__device__ __forceinline__ float ord2f(unsigned u) {
    return (u & 0x80000000u) ? __uint_as_float(u & 0x7fffffffu)
                             : __uint_as_float(~u);
}

__device__ __forceinline__ void edge_nodes(const int* __restrict__ ei, int e,
                                           int& s, int& d) {
    if (e < N_EDGES) { s = ei[e]; d = ei[N_EDGES + e]; }
    else             { s = d = e - N_EDGES; }       // appended self loops
}

__device__ __forceinline__ float lrelu(float v) {
    return v > 0.f ? v : v * NEG_SLOPE;
}

// ---------------- Kernel 1: h = x @ W via V_WMMA_F32_16X16X4_F32 ------------
// grid = N_NODES/16 blocks of 512 threads (16 waves); wave w owns N-tile w.
__global__ __launch_bounds__(512) void gat_gemm(const float* __restrict__ x,
                                                const float* __restrict__ W,
                                                float* __restrict__ h) {
    __shared__ float lA[16][IN_CH];                 // 8 KB A tile, shared by 16 waves
    const int mBase = blockIdx.x * 16;

    for (int i = threadIdx.x; i < 16 * IN_CH; i += 512) {
        int r = i >> 7, c = i & (IN_CH - 1);
        lA[r][c] = x[(mBase + r) * IN_CH + c];
    }
    __syncthreads();

    const int wave = threadIdx.x >> 5;              // 0..15 -> N tile
    const int lane = threadIdx.x & 31;
    const int lrow = lane & 15;                     // A: M row / B,C: N col index
    const int koff = (lane >> 4) * 2;               // lanes 16-31 hold K=2,3
    const int col  = wave * 16 + lrow;

    v8f acc = {};
    #pragma unroll 4
    for (int k = 0; k < IN_CH; k += 4) {
        v2f a, b;
        // A 16x4 fragment (ISA layout): lanes0-15 K=0,1 ; lanes16-31 K=2,3
        a.x = lA[lrow][k + koff];
        a.y = lA[lrow][k + koff + 1];
        // B 4x16 fragment (row striped across lanes, K-pairs mirror A)
        b.x = W[(k + koff)     * HC + col];
        b.y = W[(k + koff + 1) * HC + col];
        acc = __builtin_amdgcn_wmma_f32_16x16x4_f32(
                  false, a, false, b, (short)0, acc, false, false);
    }

    // C/D 16x16: VGPR i -> M=i (lanes 0-15) / M=8+i (lanes 16-31)
    const int mrow0 = (lane >> 4) * 8;
    #pragma unroll
    for (int i = 0; i < 8; ++i)
        h[(mBase + mrow0 + i) * HC + col] = acc[i];
}

// ---------------- Kernel 2: per-(node,head) attention logits ----------------
__global__ void gat_att(const float* __restrict__ h,
                        const float* __restrict__ att_src,
                        const float* __restrict__ att_dst,
                        float* __restrict__ asrc, float* __restrict__ adst) {
    int idx = blockIdx.x * blockDim.x + threadIdx.x;   // n*HEADS + head
    if (idx >= N_NODES * HEADS) return;
    int hd = idx & (HEADS - 1);
    const float4* hp = (const float4*)(h + (size_t)(idx >> 3) * HC + hd * OUT_CH);
    const float4* as = (const float4*)(att_src + hd * OUT_CH);
    const float4* ad = (const float4*)(att_dst + hd * OUT_CH);
    float s = 0.f, d = 0.f;
    #pragma unroll
    for (int c = 0; c < OUT_CH / 4; ++c) {
        float4 v = hp[c], a = as[c], b = ad[c];
        s += v.x * a.x + v.y * a.y + v.z * a.z + v.w * a.w;
        d += v.x * b.x + v.y * b.y + v.z * b.z + v.w * b.w;
    }
    asrc[idx] = s; adst[idx] = d;
}

// ---------------- Kernel 3: init accumulators (b128 stores) -----------------
__global__ void gat_init(float4* __restrict__ out, uint4* __restrict__ emax,
                         float4* __restrict__ esum) {
    int i = blockIdx.x * blockDim.x + threadIdx.x;
    if (i < N_NODES * HC / 4) out[i] = make_float4(0.f, 0.f, 0.f, 0.f);
    if (i < N_NODES * HEADS / 4) {
        emax[i] = make_uint4(0u, 0u, 0u, 0u);   // ord(-inf)>0; self-loops overwrite
        esum[i] = make_float4(0.f, 0.f, 0.f, 0.f);
    }
}

// ---------------- Kernel 4: segment max (1 thread / edge, 8 heads) ----------
__global__ void gat_max(const int* __restrict__ ei,
                        const float* __restrict__ asrc, const float* __restrict__ adst,
                        unsigned* __restrict__ emax) {
    int e = blockIdx.x * blockDim.x + threadIdx.x;
    if (e >= E_TOT) return;
    int s, d; edge_nodes(ei, e, s, d);
    const float4* sp = (const float4*)(asrc + (size_t)s * HEADS);
    const float4* dp = (const float4*)(adst + (size_t)d * HEADS);
    float4 s0 = sp[0], s1 = sp[1], d0 = dp[0], d1 = dp[1];
    float v[HEADS] = { s0.x + d0.x, s0.y + d0.y, s0.z + d0.z, s0.w + d0.w,
                       s1.x + d1.x, s1.y + d1.y, s1.z + d1.z, s1.w + d1.w };
    unsigned* mp = emax + (size_t)d * HEADS;
    #pragma unroll
    for (int i = 0; i < HEADS; ++i)
        atomicMax(mp + i, f2ord(lrelu(v[i])));
}

// ---------------- Kernel 5: segment sum of exp (1 thread / edge) ------------
__global__ void gat_sum(const int* __restrict__ ei,
                        const float* __restrict__ asrc, const float* __restrict__ adst,
                        const unsigned* __restrict__ emax, float* __restrict__ esum) {
    int e = blockIdx.x * blockDim.x + threadIdx.x;
    if (e >= E_TOT) return;
    int s, d; edge_nodes(ei, e, s, d);
    const float4* sp = (const float4*)(asrc + (size_t)s * HEADS);
    const float4* dp = (const float4*)(adst + (size_t)d * HEADS);
    const uint4*  xp = (const uint4*)(emax + (size_t)d * HEADS);
    float4 s0 = sp[0], s1 = sp[1], d0 = dp[0], d1 = dp[1];
    uint4  m0 = xp[0], m1 = xp[1];
    float v[HEADS] = { s0.x + d0.x, s0.y + d0.y, s0.z + d0.z, s0.w + d0.w,
                       s1.x + d1.x, s1.y + d1.y, s1.z + d1.z, s1.w + d1.w };
    unsigned m[HEADS] = { m0.x, m0.y, m0.z, m0.w, m1.x, m1.y, m1.z, m1.w };
    float* op = esum + (size_t)d * HEADS;
    #pragma unroll
    for (int i = 0; i < HEADS; ++i)
        atomicAdd(op + i, expf(lrelu(v[i]) - ord2f(m[i])));
}

// ---------------- Kernel 6: weighted message aggregation --------------------
// 4 edges / 256-thread block; 64 threads per edge, b128 gather of h row.
// h and out both live in the 192MB L2 -> gathers/atomics resolve on-chip.
__global__ __launch_bounds__(256) void gat_agg(const int* __restrict__ ei,
                                               const float* __restrict__ h,
                                               const float* __restrict__ asrc,
                                               const float* __restrict__ adst,
                                               const unsigned* __restrict__ emax,
                                               const float* __restrict__ esum,
                                               float* __restrict__ out) {
    int t = threadIdx.x;
    int e = blockIdx.x * 4 + (t >> 6);
    if (e >= E_TOT) return;
    int ch = (t & 63) * 4;                         // channel base, 0..252
    int hd = ch >> 5;                              // 4 channels share one head
    int s, d; edge_nodes(ei, e, s, d);
    float v = lrelu(asrc[(size_t)s * HEADS + hd] + adst[(size_t)d * HEADS + hd]);
    float m = ord2f(emax[(size_t)d * HEADS + hd]);
    float alpha = expf(v - m) / (esum[(size_t)d * HEADS + hd] + 1e-16f);
    float4 hv = *(const float4*)(h + (size_t)s * HC + ch);
    float* op = out + (size_t)d * HC + ch;
    atomicAdd(op + 0, alpha * hv.x);
    atomicAdd(op + 1, alpha * hv.y);
    atomicAdd(op + 2, alpha * hv.z);
    atomicAdd(op + 3, alpha * hv.w);
}

// ---------------- Kernel 7: bias + ReLU (b128) ------------------------------
__global__ void gat_final(float4* __restrict__ out, const float4* __restrict__ bias) {
    int i = blockIdx.x * blockDim.x + threadIdx.x;
    if (i >= N_NODES * HC / 4) return;
    float4 v = out[i];
    float4 b = bias[i & (HC / 4 - 1)];
    v.x = fmaxf(v.x + b.x, 0.f);
    v.y = fmaxf(v.y + b.y, 0.f);
    v.z = fmaxf(v.z + b.z, 0.f);
    v.w = fmaxf(v.w + b.w, 0.f);
    out[i] = v;
}

extern "C" void kernel_launch(void* const* d_in, const int* in_sizes, int n_in,
                              void* d_out, int out_size, void* d_ws, size_t ws_size,
                              hipStream_t stream) {
    const float* x        = (const float*)d_in[0];   // [N, 128]
    const int*   ei       = (const int*)  d_in[1];   // [2, E]
    const float* W        = (const float*)d_in[2];   // [128, 256]
    const float* att_src  = (const float*)d_in[3];   // [8, 32]
    const float* att_dst  = (const float*)d_in[4];   // [8, 32]
    const float* bias     = (const float*)d_in[5];   // [256]
    float* out = (float*)d_out;                      // [N, 256]

    char* ws = (char*)d_ws;
    float*    h    = (float*)(ws);                                        // 51.2 MB
    float*    asrc = (float*)(ws + (size_t)N_NODES * HC * 4);             // 1.6 MB
    float*    adst = asrc + (size_t)N_NODES * HEADS;                      // 1.6 MB
    unsigned* emax = (unsigned*)(adst + (size_t)N_NODES * HEADS);         // 1.6 MB
    float*    esum = (float*)(emax + (size_t)N_NODES * HEADS);            // 1.6 MB

    const int nh   = N_NODES * HEADS;
    const int nhc  = N_NODES * HC;

    gat_init <<<(nhc / 4 + 255) / 256, 256, 0, stream>>>(
        (float4*)out, (uint4*)emax, (float4*)esum);
    gat_gemm <<<N_NODES / 16, 512, 0, stream>>>(x, W, h);
    gat_att  <<<(nh + 255) / 256, 256, 0, stream>>>(h, att_src, att_dst, asrc, adst);
    gat_max  <<<(E_TOT + 255) / 256, 256, 0, stream>>>(ei, asrc, adst, emax);
    gat_sum  <<<(E_TOT + 255) / 256, 256, 0, stream>>>(ei, asrc, adst, emax, esum);
    gat_agg  <<<(E_TOT + 3) / 4, 256, 0, stream>>>(ei, h, asrc, adst, emax, esum, out);
    gat_final<<<(nhc / 4 + 255) / 256, 256, 0, stream>>>((float4*)out, (const float4*)bias);
}